// CumulativeGroupNorm_69698729279770
// MI455X (gfx1250) — compile-verified
//
#include <hip/hip_runtime.h>

// CumulativeGroupNorm on MI455X (gfx1250), fp32 in/out.
// Shapes fixed by reference: B=4, T=2048, F=128, C=128  (N = F*C = 16384).
//
// Strategy (bandwidth-bound, ~1-1.5 GB HBM traffic):
//   per batch b (128 MB slab, fits MI455X's 192 MB L2):
//     1) cgn_reduce:  per-t sum / sumsq via V_WMMA_F32_16X16X4_F32 (all-ones B)
//     2) cgn_scan:    length-2048 scan -> mean[t], invstd[t]
//     3) cgn_norm:    (x - mean)*invstd*weight, NT stores so x_b stays in L2
//   Batch-sequenced launches keep x_b L2-resident between pass 1 and pass 3.

typedef float f4  __attribute__((ext_vector_type(4)));
typedef float v2f __attribute__((ext_vector_type(2)));
typedef float v8f __attribute__((ext_vector_type(8)));

#define B_DIM 4
#define T_DIM 2048
#define N_RED 16384      // F*C elements reduced per timestep
#define C_DIM 128
#define CGN_EPS 1e-3f

// ---------------------------------------------------------------------------
// Pass 1: one block per timestep; 256 threads x 16 float4 = 16384 floats.
// Reduction runs on the matrix pipe: D = A x ones + C accumulates row sums.
// With B == all-ones, every element of A is counted exactly once regardless
// of the A VGPR layout, so any lane->slot mapping is correct.
// ---------------------------------------------------------------------------
__global__ __launch_bounds__(256) void cgn_reduce(const float* __restrict__ xb,
                                                  float* __restrict__ sums,
                                                  float* __restrict__ sumsq) {
  const int t   = blockIdx.x;
  const int tid = threadIdx.x;
  const f4* __restrict__ p = (const f4*)xb + (size_t)t * (N_RED / 4);

  v8f csum = {};
  v8f csq  = {};
  const v2f ones = {1.0f, 1.0f};

#pragma unroll
  for (int i = 0; i < 16; ++i) {
    f4 v = p[tid + i * 256];                       // global_load_b128, coalesced
    v2f a0 = {v.x, v.y};
    v2f a1 = {v.z, v.w};
    v2f q0 = a0 * a0;
    v2f q1 = a1 * a1;
    // v_wmma_f32_16x16x4_f32: D = A(16x4) x ones(4x16) + C  -> per-row sums
    csum = __builtin_amdgcn_wmma_f32_16x16x4_f32(false, a0, false, ones,
                                                 (short)0, csum, false, false);
    csum = __builtin_amdgcn_wmma_f32_16x16x4_f32(false, a1, false, ones,
                                                 (short)0, csum, false, false);
    csq  = __builtin_amdgcn_wmma_f32_16x16x4_f32(false, q0, false, ones,
                                                 (short)0, csq,  false, false);
    csq  = __builtin_amdgcn_wmma_f32_16x16x4_f32(false, q1, false, ones,
                                                 (short)0, csq,  false, false);
  }

  // C layout: lanes 0-15 hold rows 0-7 (one row per VGPR), lanes 16-31 rows
  // 8-15; all 16 columns identical (B all-ones). Horizontal-sum the 8 slots,
  // then fold the two lane halves -> grand total in every lane.
  float s = csum[0] + csum[1] + csum[2] + csum[3] +
            csum[4] + csum[5] + csum[6] + csum[7];
  float q = csq[0] + csq[1] + csq[2] + csq[3] +
            csq[4] + csq[5] + csq[6] + csq[7];
  s += __shfl_xor(s, 16, 32);
  q += __shfl_xor(q, 16, 32);

  __shared__ float reds[8];
  __shared__ float redq[8];
  const int wave = tid >> 5;
  const int lane = tid & 31;
  if (lane == 0) { reds[wave] = s; redq[wave] = q; }
  __syncthreads();
  if (tid == 0) {
    float ts = 0.0f, tq = 0.0f;
#pragma unroll
    for (int w = 0; w < 8; ++w) { ts += reds[w]; tq += redq[w]; }
    sums[t]  = ts;
    sumsq[t] = tq;
  }
}

// ---------------------------------------------------------------------------
// Pass 2: one block per batch. 256 threads x 8 timesteps. Two inclusive scans
// (cum_sum, then cum_sq_diff via the expanded form). Cost is negligible.
// ---------------------------------------------------------------------------
__global__ __launch_bounds__(256) void cgn_scan(const float* __restrict__ sums,
                                                const float* __restrict__ sumsq,
                                                float* __restrict__ mean,
                                                float* __restrict__ invstd) {
  __shared__ float sdata[256];
  const int k = threadIdx.x;

  float s_orig[8], incl[8];
  float run = 0.0f;
#pragma unroll
  for (int i = 0; i < 8; ++i) {
    s_orig[i] = sums[k * 8 + i];
    run += s_orig[i];
    incl[i] = run;
  }

  // Block-wide inclusive scan of per-thread totals (Hillis-Steele).
  sdata[k] = run;
  __syncthreads();
  for (int off = 1; off < 256; off <<= 1) {
    float v = (k >= off) ? sdata[k - off] : 0.0f;
    __syncthreads();
    sdata[k] += v;
    __syncthreads();
  }
  const float excl = sdata[k] - run;

  float d[8];
#pragma unroll
  for (int i = 0; i < 8; ++i) {
    const int   t   = k * 8 + i;
    const float cs  = excl + incl[i];
    const float cnt = (float)(t + 1) * (float)N_RED;
    const float m   = cs / cnt;
    mean[t] = m;
    // sum_{f,c}(x - m)^2 = sumsq - 2*m*sum + N*m^2
    d[i] = sumsq[t] - 2.0f * m * s_orig[i] + (float)N_RED * m * m;
  }

  float run2 = 0.0f, incl2[8];
#pragma unroll
  for (int i = 0; i < 8; ++i) { run2 += d[i]; incl2[i] = run2; }

  __syncthreads();
  sdata[k] = run2;
  __syncthreads();
  for (int off = 1; off < 256; off <<= 1) {
    float v = (k >= off) ? sdata[k - off] : 0.0f;
    __syncthreads();
    sdata[k] += v;
    __syncthreads();
  }
  const float excl2 = sdata[k] - run2;

#pragma unroll
  for (int i = 0; i < 8; ++i) {
    const int   t   = k * 8 + i;
    const float cnt = (float)(t + 1) * (float)N_RED;
    const float var = (excl2 + incl2[i]) / cnt;
    invstd[t] = rsqrtf(var + CGN_EPS);
  }
}

// ---------------------------------------------------------------------------
// Pass 3: elementwise normalize. x loads hit L2 (warm from pass 1); output is
// written with non-temporal stores so the out-stream does not evict x_b.
// ---------------------------------------------------------------------------
__global__ __launch_bounds__(256) void cgn_norm(const float* __restrict__ xb,
                                                const float* __restrict__ w,
                                                const float* __restrict__ mean,
                                                const float* __restrict__ invstd,
                                                float* __restrict__ outb) {
  const size_t g = (size_t)blockIdx.x * 256 + threadIdx.x;  // float4 index
  const int    t = (int)(g >> 12);                          // 4096 f4 per t

  f4 xv = ((const f4*)xb)[g];
  f4 wv = ((const f4*)w)[g & 31];                           // C=128 -> 32 f4
  const float m  = mean[t];
  const float is = invstd[t];

  f4 o = (xv - m) * is * wv;
  __builtin_nontemporal_store(o, (f4*)outb + g);            // store TH=NT
}

// ---------------------------------------------------------------------------
extern "C" void kernel_launch(void* const* d_in, const int* in_sizes, int n_in,
                              void* d_out, int out_size, void* d_ws, size_t ws_size,
                              hipStream_t stream) {
  const float* x   = (const float*)d_in[0];   // [B, T, F, C] fp32
  const float* w   = (const float*)d_in[1];   // [C] fp32
  float*       out = (float*)d_out;

  float* ws     = (float*)d_ws;               // 4*T floats = 32 KB scratch
  float* sums   = ws;
  float* sumsq  = ws + T_DIM;
  float* mean   = ws + 2 * T_DIM;
  float* invstd = ws + 3 * T_DIM;

  const size_t batch_elems = (size_t)T_DIM * N_RED;

  // Batch-sequenced so each 128 MB slab stays resident in the 192 MB L2
  // between the reduce pass and the normalize pass.
  for (int b = 0; b < B_DIM; ++b) {
    const float* xb = x + (size_t)b * batch_elems;
    float*       ob = out + (size_t)b * batch_elems;

    cgn_reduce<<<T_DIM, 256, 0, stream>>>(xb, sums, sumsq);
    cgn_scan<<<1, 256, 0, stream>>>(sums, sumsq, mean, invstd);
    cgn_norm<<<(int)(batch_elems / 4 / 256), 256, 0, stream>>>(xb, w, mean,
                                                               invstd, ob);
  }
}